// TuckerDecomposition_48928267436479
// MI455X (gfx1250) — compile-verified
//
#include <hip/hip_runtime.h>

// Tucker contraction for MI455X (gfx1250), wave32.
// out[a,s,t] = sum_{f,e,g} act[a,f]*state[s,e]*core[f,e,g]*state[t,g]
// A=64, S=T=1024, E=G=64, F=32. Output 256MB fp32 -> ~11us HBM floor @23.3TB/s.
// Stage-3 GEMMs (8.6 GFLOP) run on V_WMMA_F32_16X16X4_F32 (exact fp32 math).

typedef __attribute__((ext_vector_type(2))) float v2f;
typedef __attribute__((ext_vector_type(8))) float v8f;

#define F_DIM 32

// ---------------------------------------------------------------------------
// Kernel 1: c2[a,e,g] = sum_f act[a,f] * core[f,e,g]   (64x64x64 fp32, 1 MB)
// 16.7 MFLOP total -> scalar VALU is fine.
// ---------------------------------------------------------------------------
__global__ void tucker_c2_kernel(const float* __restrict__ act,
                                 const float* __restrict__ core,
                                 float* __restrict__ c2) {
    int idx = blockIdx.x * blockDim.x + threadIdx.x;   // 0 .. 262143
    int g = idx & 63;
    int e = (idx >> 6) & 63;
    int a = idx >> 12;
    float acc = 0.0f;
    #pragma unroll
    for (int f = 0; f < F_DIM; ++f)
        acc = fmaf(act[a * F_DIM + f], core[(f * 64 + e) * 64 + g], acc);
    c2[idx] = acc;
}

// ---------------------------------------------------------------------------
// Kernel 2 (fused stages 2+3):
// Block = (s-strip of 128 rows, one a). 8 waves.
// Phase 1: t2s[sl,g] = sum_e state[s,e]*c2[a,e,g] -> 32KB LDS tile (WMMA).
// Phase 2: out[a,s,t] = sum_g t2s[sl,g]*state[t,g]; wave w owns t in
//          [128w,128w+128); 8 s-tile accumulators reuse each B fragment.
// ---------------------------------------------------------------------------
__global__ __launch_bounds__(256)
void tucker_main_kernel(const float* __restrict__ state,   // [1024][64]
                        const float* __restrict__ c2,      // [64][64][64]
                        float* __restrict__ out) {         // [64][1024][1024]
    __shared__ float t2s[128 * 64];                        // 32 KB of 320 KB

    const int sblk = blockIdx.x;          // 0..7
    const int a    = blockIdx.y;          // 0..63
    const int lane = threadIdx.x & 31;
    const int wave = threadIdx.x >> 5;    // 0..7
    const int n    = lane & 15;           // N col / A row within half-wave
    const int h    = lane >> 4;           // half-wave select
    const int s_base = sblk * 128;

    // ---- Phase 1: wave w computes t2s rows [16w, 16w+16), all 64 g cols ----
    {
        v8f acc[4] = {};                                  // 4 g-tiles of 16
        const float* A_row  = state + (s_base + 16 * wave + n) * 64; // m = n
        const float* B_base = c2 + a * 64 * 64;
        for (int ks = 0; ks < 16; ++ks) {                 // K = 64, 4 per WMMA
            const int k0 = ks * 4 + 2 * h;
            v2f afrag = *(const v2f*)(A_row + k0);        // state[s, e..e+1]
            #pragma unroll
            for (int gt = 0; gt < 4; ++gt) {
                v2f bfrag;
                bfrag.x = B_base[(k0    ) * 64 + gt * 16 + n];  // c2[a,e  ,g]
                bfrag.y = B_base[(k0 + 1) * 64 + gt * 16 + n];  // c2[a,e+1,g]
                acc[gt] = __builtin_amdgcn_wmma_f32_16x16x4_f32(
                    false, afrag, false, bfrag, (short)0, acc[gt], false, false);
            }
        }
        #pragma unroll
        for (int gt = 0; gt < 4; ++gt)
            #pragma unroll
            for (int r = 0; r < 8; ++r)                   // D: row r+8h, col n
                t2s[(16 * wave + r + 8 * h) * 64 + gt * 16 + n] = acc[gt][r];
    }
    __syncthreads();

    // ---- Phase 2: wave w owns t in [128w, 128w+128) ----
    const int t_wave = 128 * wave;
    for (int tt = 0; tt < 8; ++tt) {
        const int t0 = t_wave + 16 * tt;
        v8f acc[8] = {};                                  // 8 s-tiles
        const float* B_row = state + (t0 + n) * 64;       // state[t, g..g+1]
        for (int ks = 0; ks < 16; ++ks) {
            const int k0 = ks * 4 + 2 * h;
            v2f bfrag = *(const v2f*)(B_row + k0);        // 1 global load
            #pragma unroll
            for (int st = 0; st < 8; ++st) {              // 8 WMMAs reuse it
                v2f afrag = *(const v2f*)(&t2s[(16 * st + n) * 64 + k0]);
                acc[st] = __builtin_amdgcn_wmma_f32_16x16x4_f32(
                    false, afrag, false, bfrag, (short)0, acc[st], false, false);
            }
        }
        // Streamed 256MB output: non-temporal so L2 keeps state/c2 resident.
        #pragma unroll
        for (int st = 0; st < 8; ++st) {
            const unsigned srow = (unsigned)(s_base + 16 * st + 8 * h);
            #pragma unroll
            for (int r = 0; r < 8; ++r) {
                float* p = out + (((unsigned)a << 20) |
                                  ((srow + r) << 10) | (unsigned)(t0 + n));
                __builtin_nontemporal_store(acc[st][r], p);
            }
        }
    }
}

// ---------------------------------------------------------------------------
extern "C" void kernel_launch(void* const* d_in, const int* in_sizes, int n_in,
                              void* d_out, int out_size, void* d_ws, size_t ws_size,
                              hipStream_t stream) {
    const float* state = (const float*)d_in[0];   // (1,1024,64)
    const float* act   = (const float*)d_in[1];   // (1,64,32)
    const float* core  = (const float*)d_in[2];   // (1,32,64,64)
    float* out = (float*)d_out;                   // (1,64,1024,1024)
    float* c2  = (float*)d_ws;                    // 1 MB scratch

    tucker_c2_kernel<<<1024, 256, 0, stream>>>(act, core, c2);
    dim3 grid(8, 64);                             // (s-strips, a)
    tucker_main_kernel<<<grid, 256, 0, stream>>>(state, c2, out);
}